// LooseMatchAssignment_8452495639155
// MI455X (gfx1250) — compile-verified
//
#include <hip/hip_runtime.h>
#include <math.h>

// -------- problem constants (from reference setup_inputs) --------
#define BATCH   8
#define MROWS   2048
#define NCOLS   2048
#define INDIM   256
#define PDIM    2049            // probs is (b, m+1, n+1)

typedef float v2f __attribute__((ext_vector_type(2)));
typedef float v8f __attribute__((ext_vector_type(8)));

// -------- workspace layout (float offsets) --------
// Bbuf packed per lane: Bbuf[it*64 + lane*2 + v]  (v = VGPR index 0/1)
// so each lane fetches its two B VGPR values with one b64 load.
#define WS_BBUF0   0            // 4096 floats (side 0: v0 + Wm columns)
#define WS_BBUF1   4096         // 4096 floats (side 1: v1 + Wm columns)
#define WS_CCONST  8192         // c0, c1
#define WS_S0      8704         // 16384 each below (BATCH*MROWS)
#define WS_L0      (WS_S0 + BATCH*MROWS)
#define WS_E0      (WS_L0 + BATCH*MROWS)
#define WS_S1      (WS_E0 + BATCH*MROWS)
#define WS_L1      (WS_S1 + BATCH*MROWS)
#define WS_E1      (WS_L1 + BATCH*MROWS)

// Fast, numerically stable log_sigmoid using HW transcendentals
// (v_exp_f32 + v_log_f32).  Abs error vs log1pf path < ~1e-7.
__device__ __forceinline__ float lsig(float x) {
    return fminf(x, 0.0f) - __logf(1.0f + __expf(-fabsf(x)));
}

// ---------------------------------------------------------------------------
// Kernel 1: fold weights.  v0[i] = sum_d W0[d,i]*Wc[d]; v1[i] = sum_d W1[d,i]*Wc[128+d]
// c0 = b0.Wc[:128]; c1 = b1.Wc[128:].  Then pack WMMA B operand buffers.
// B is 4x16 (KxN); VGPR0 holds K rows {0 (lanes0-15), 2 (lanes16-31)},
// VGPR1 holds {1, 3}.  Column 0 = v, column 1 = Wm, columns 2..15 = 0.
// Packed layout: Bbuf[it*64 + lane*2 + vgpr].
// ---------------------------------------------------------------------------
__global__ void prep_kernel(const float* __restrict__ W0, const float* __restrict__ b0,
                            const float* __restrict__ W1, const float* __restrict__ b1,
                            const float* __restrict__ Wm, const float* __restrict__ Wc,
                            float* __restrict__ ws) {
    __shared__ float sv0[INDIM];
    __shared__ float sv1[INDIM];
    int i = threadIdx.x;              // 0..255
    float a0 = 0.0f, a1 = 0.0f;
    for (int d = 0; d < 128; ++d) {
        a0 += W0[d * INDIM + i] * Wc[d];
        a1 += W1[d * INDIM + i] * Wc[128 + d];
    }
    sv0[i] = a0;
    sv1[i] = a1;
    if (i == 0) {
        float c0 = 0.0f, c1 = 0.0f;
        for (int d = 0; d < 128; ++d) {
            c0 += b0[d] * Wc[d];
            c1 += b1[d] * Wc[128 + d];
        }
        ws[WS_CCONST + 0] = c0;
        ws[WS_CCONST + 1] = c1;
    }
    __syncthreads();
    for (int t = i; t < 4096; t += 256) {
        int it   = t >> 6;
        int r    = t & 63;
        int lane = r >> 1;              // 0..31
        int vg   = r & 1;               // VGPR 0/1
        int jcol = lane & 15;
        int k    = it * 4 + ((lane < 16) ? vg : vg + 2);
        float wmk = Wm[k];
        ws[WS_BBUF0 + t] = (jcol == 0) ? sv0[k] : ((jcol == 1) ? wmk : 0.0f);
        ws[WS_BBUF1 + t] = (jcol == 0) ? sv1[k] : ((jcol == 1) ? wmk : 0.0f);
    }
}

// ---------------------------------------------------------------------------
// Kernel 2: per-row stats via V_WMMA_F32_16X16X4_F32.
// One wave handles 16 rows; 64 chained WMMAs accumulate the 256-long dots.
// A (16x4 f32): lanes 0-15 = rows M, VGPR0/1 = K 0/1; lanes 16-31 same rows, K 2/3.
// The hi-lane K offset (+2) is folded into the base pointer, so each
// iteration is exactly one b64 A-load + one b64 B-load + one WMMA.
// Output C: VGPR r -> (M=r, N=lane) lanes 0-15, (M=8+r, N=lane-16) lanes 16-31.
// Column N=0 is the s-dot, N=1 is the z-dot.
// ---------------------------------------------------------------------------
__global__ void rowstats_kernel(const float* __restrict__ desc0,
                                const float* __restrict__ desc1,
                                const float* __restrict__ bm,
                                float* __restrict__ ws) {
    int wave = (blockIdx.x * blockDim.x + threadIdx.x) >> 5;  // 0..2047
    int lane = threadIdx.x & 31;
    int side = wave >> 10;          // 1024 tiles per side
    int tile = wave & 1023;

    const float* desc = side ? desc1 : desc0;
    const float* Bbuf = ws + (side ? WS_BBUF1 : WS_BBUF0);
    float cc  = ws[WS_CCONST + side];
    float bmv = bm[0];
    float* S = ws + (side ? WS_S1 : WS_S0);
    float* L = ws + (side ? WS_L1 : WS_L0);
    float* E = ws + (side ? WS_E1 : WS_E0);

    int  mrow = lane & 15;
    bool hi   = lane >= 16;
    // A element pair for this lane at iteration it: desc[row][it*4 + (hi?2:0) .. +1]
    const v2f* rowp = reinterpret_cast<const v2f*>(
        desc + (size_t)(tile * 16 + mrow) * INDIM + (hi ? 2 : 0));
    const v2f* bp = reinterpret_cast<const v2f*>(Bbuf) + lane;

    v8f acc = {0.0f, 0.0f, 0.0f, 0.0f, 0.0f, 0.0f, 0.0f, 0.0f};
    #pragma unroll 4
    for (int it = 0; it < 64; ++it) {
        v2f A = rowp[it * 2];           // global_load_b64
        v2f B = bp[it * 32];            // global_load_b64
        acc = __builtin_amdgcn_wmma_f32_16x16x4_f32(
                  /*neg_a=*/false, A, /*neg_b=*/false, B,
                  /*c_mod=*/(short)0, acc, /*reuse_a=*/false, /*reuse_b=*/false);
    }

    int ncol  = lane & 15;
    int gbase = tile * 16 + (hi ? 8 : 0);   // flat row index over (b*2048)
    if (ncol == 0) {
        #pragma unroll
        for (int r = 0; r < 8; ++r) S[gbase + r] = acc[r] + cc;
    } else if (ncol == 1) {
        #pragma unroll
        for (int r = 0; r < 8; ++r) {
            float z = acc[r] + bmv;
            L[gbase + r] = lsig(z);
            E[gbase + r] = lsig(-z);
        }
    }
}

// ---------------------------------------------------------------------------
// Kernel 3: the bandwidth-bound broadcast.  Block = 256 columns x 16 rows.
// Non-temporal stores: 268 MB of output exceeds the 192 MB L2.
// ---------------------------------------------------------------------------
#define ROWS_PER_BLOCK 16
__global__ void broadcast_kernel(const float* __restrict__ ws,
                                 const float* __restrict__ bc,
                                 float* __restrict__ probs,
                                 float* __restrict__ corres) {
    int j  = blockIdx.x * blockDim.x + threadIdx.x;   // 0..2047
    int b  = blockIdx.z;
    int i0 = blockIdx.y * ROWS_PER_BLOCK;

    const float* S0 = ws + WS_S0 + b * MROWS;
    const float* L0 = ws + WS_L0 + b * MROWS;
    float s1  = ws[WS_S1 + b * NCOLS + j];
    float l1  = ws[WS_L1 + b * NCOLS + j];
    float bcv = bc[0];

    float* prow = probs  + (size_t)b * PDIM * PDIM + (size_t)i0 * PDIM + j;
    float* crow = corres + (size_t)b * MROWS * NCOLS + (size_t)i0 * NCOLS + j;

    #pragma unroll 4
    for (int r = 0; r < ROWS_PER_BLOCK; ++r) {
        float s0 = S0[i0 + r];                 // uniform -> scalar load
        float l0 = L0[i0 + r];
        float c  = s0 + s1 + bcv;
        __builtin_nontemporal_store(c, crow);
        __builtin_nontemporal_store(lsig(c) + l0 + l1, prow);
        prow += PDIM;
        crow += NCOLS;
    }
}

// ---------------------------------------------------------------------------
// Kernel 4: borders.  probs[b, i, N] = ls(-z0), probs[b, M, j] = ls(-z1),
// probs[b, M, N] = 0.
// ---------------------------------------------------------------------------
__global__ void border_kernel(const float* __restrict__ ws, float* __restrict__ probs) {
    int t = blockIdx.x * blockDim.x + threadIdx.x;
    if (t >= BATCH * PDIM) return;
    int b = t / PDIM;
    int q = t % PDIM;
    float* pb = probs + (size_t)b * PDIM * PDIM;
    if (q < NCOLS) {
        pb[(size_t)q * PDIM + NCOLS]     = ws[WS_E0 + b * MROWS + q];
        pb[(size_t)MROWS * PDIM + q]     = ws[WS_E1 + b * NCOLS + q];
    } else {
        pb[(size_t)MROWS * PDIM + NCOLS] = 0.0f;
    }
}

// ---------------------------------------------------------------------------
extern "C" void kernel_launch(void* const* d_in, const int* in_sizes, int n_in,
                              void* d_out, int out_size, void* d_ws, size_t ws_size,
                              hipStream_t stream) {
    const float* desc0 = (const float*)d_in[0];
    const float* desc1 = (const float*)d_in[1];
    const float* W0    = (const float*)d_in[2];
    const float* b0    = (const float*)d_in[3];
    const float* W1    = (const float*)d_in[4];
    const float* b1    = (const float*)d_in[5];
    const float* Wm    = (const float*)d_in[6];
    const float* bm    = (const float*)d_in[7];
    const float* Wc    = (const float*)d_in[8];
    const float* bc    = (const float*)d_in[9];

    float* probs  = (float*)d_out;                                  // (8,2049,2049)
    float* corres = probs + (size_t)BATCH * PDIM * PDIM;            // (8,2048,2048)
    float* ws     = (float*)d_ws;

    prep_kernel<<<1, 256, 0, stream>>>(W0, b0, W1, b1, Wm, Wc, ws);

    // 2048 waves total (1024 tiles per side), 8 waves per block
    rowstats_kernel<<<256, 256, 0, stream>>>(desc0, desc1, bm, ws);

    dim3 grid(NCOLS / 256, MROWS / ROWS_PER_BLOCK, BATCH);
    broadcast_kernel<<<grid, 256, 0, stream>>>(ws, bc, probs, corres);

    border_kernel<<<(BATCH * PDIM + 255) / 256, 256, 0, stream>>>(ws, probs);
}